// SkeletonModel32_34445637714246
// MI455X (gfx1250) — compile-verified
//
#include <hip/hip_runtime.h>

#ifndef __has_builtin
#define __has_builtin(x) 0
#endif

#if __has_builtin(__builtin_amdgcn_global_load_async_to_lds_b128) && \
    __has_builtin(__builtin_amdgcn_global_store_async_from_lds_b32)
#define USE_ASYNC_LDS 1
#else
#define USE_ASYNC_LDS 0
#endif

#define FK_BLOCK 128
#define FK_XF 99   // floats per input row
#define FK_PF 75   // floats per output row

// Async-LDS builtins take element-typed pointers in AS(1) (global) / AS(3) (LDS).
typedef int fk_v4i __attribute__((ext_vector_type(4)));
typedef fk_v4i __attribute__((address_space(1)))* fk_gv4p;
typedef fk_v4i __attribute__((address_space(3)))* fk_lv4p;
typedef int __attribute__((address_space(1)))* fk_gi32p;
typedef int __attribute__((address_space(3)))* fk_li32p;

__device__ __forceinline__ void fk_rodrigues(float ax, float ay, float az, float R[9]) {
  float t2 = ax * ax + ay * ay + az * az;
  float th = sqrtf(t2 + 1e-12f);
  float inv = 1.0f / th;
  float kx = ax * inv, ky = ay * inv, kz = az * inv;
  float c = cosf(th);
  float s = sinf(th);
  float oc = 1.0f - c;
  float ocx = oc * kx, ocy = oc * ky, ocz = oc * kz;
  R[0] = c + ocx * kx;      R[1] = ocx * ky - s * kz; R[2] = ocx * kz + s * ky;
  R[3] = ocx * ky + s * kz; R[4] = c + ocy * ky;      R[5] = ocy * kz - s * kx;
  R[6] = ocx * kz - s * ky; R[7] = ocy * kz + s * kx; R[8] = c + ocz * kz;
}

// Computes pos for one sample; pos[3j+c] is written as soon as joint j is done.
// `pos` MAY ALIAS `in` (main kernel passes the same LDS row): safe because the
// angle triple indices kTrip[j] >= 3(j+1), so slot [3j,3j+2] is consumed before
// iteration j writes it. No __restrict__ between in/pos on purpose.
__device__ __forceinline__ void fk_row(const float* in, const float* __restrict__ off,
                                       float* pos) {
  constexpr int kPar[25]  = {-1,0,1,2,3,0,5,6,7,0,9,10,11,10,13,14,15,16,16,10,19,20,21,22,22};
  constexpr int kTrip[25] = {3,6,9,12,15,21,24,27,30,36,39,42,45,51,54,57,60,63,69,75,78,81,84,87,93};
  float A[25][9];
#pragma unroll
  for (int j = 0; j < 25; ++j) {
    const int t = kTrip[j];
    float R[9];
    fk_rodrigues(in[t + 0], in[t + 1], in[t + 2], R);
    if (j == 0) {
      float p0 = off[0] + in[0], p1 = off[1] + in[1], p2 = off[2] + in[2];
      pos[0] = p0; pos[1] = p1; pos[2] = p2;
#pragma unroll
      for (int q = 0; q < 9; ++q) A[0][q] = R[q];
    } else {
      const int p = kPar[j];
      const float o0 = off[3 * j + 0], o1 = off[3 * j + 1], o2 = off[3 * j + 2];
      const float pp0 = pos[3 * p + 0], pp1 = pos[3 * p + 1], pp2 = pos[3 * p + 2];
      float q0, q1, q2;
      {
        q0 = o0 * A[p][0] + o1 * A[p][3] + o2 * A[p][6] + pp0;
        q1 = o0 * A[p][1] + o1 * A[p][4] + o2 * A[p][7] + pp1;
        q2 = o0 * A[p][2] + o1 * A[p][5] + o2 * A[p][8] + pp2;
      }
#pragma unroll
      for (int c = 0; c < 3; ++c) {
        A[j][c]     = R[0] * A[p][c] + R[1] * A[p][3 + c] + R[2] * A[p][6 + c];
        A[j][3 + c] = R[3] * A[p][c] + R[4] * A[p][3 + c] + R[5] * A[p][6 + c];
        A[j][6 + c] = R[6] * A[p][c] + R[7] * A[p][3 + c] + R[8] * A[p][6 + c];
      }
      pos[3 * j + 0] = q0; pos[3 * j + 1] = q1; pos[3 * j + 2] = q2;
    }
  }
}

__global__ __launch_bounds__(FK_BLOCK) void fk_kernel(const float* __restrict__ x,
                                                      const float* __restrict__ off,
                                                      float* __restrict__ out) {
  __shared__ float lds[FK_BLOCK * FK_XF];  // 50688 B; positions written in place
  const int tid = threadIdx.x;
  const long long blk = blockIdx.x;
  const float* gx = x + blk * (long long)(FK_BLOCK * FK_XF);
  float* gout = out + blk * (long long)(FK_BLOCK * FK_PF);

  // ---- Phase 1: stage x tile into LDS, coalesced b128 per lane ----
  constexpr int NV_IN = FK_BLOCK * FK_XF / 4;  // 3168 float4 chunks
#if USE_ASYNC_LDS
  for (int i = tid; i < NV_IN; i += FK_BLOCK) {
    __builtin_amdgcn_global_load_async_to_lds_b128(
        (fk_gv4p)(gx + 4 * i), (fk_lv4p)(&lds[4 * i]), 0, 0);
  }
#if __has_builtin(__builtin_amdgcn_s_wait_asynccnt)
  __builtin_amdgcn_s_wait_asynccnt(0);
#else
  asm volatile("s_wait_asynccnt 0" ::: "memory");
#endif
#else
  for (int i = tid; i < NV_IN; i += FK_BLOCK)
    ((float4*)lds)[i] = ((const float4*)gx)[i];
#endif
  __syncthreads();

  // ---- Phase 2: FK chain; pos[j] overwrites consumed slots of this thread's
  //      own LDS row (row is private to the thread; stride 99 is odd =>
  //      bank-conflict-free across the wave) ----
  float* row = &lds[tid * FK_XF];
  fk_row(row, off, row);
  __syncthreads();

  // ---- Phase 3: coalesced global stores; LDS side is strided (row*99 + c),
  //      global side is linear — per-lane async b32 handles the remap ----
  constexpr int NOUT = FK_BLOCK * FK_PF;  // 9600 floats
#if USE_ASYNC_LDS
  for (int i = tid; i < NOUT; i += FK_BLOCK) {
    const int r = i / FK_PF;            // constant divisor -> mulhi
    const int c = i - r * FK_PF;
    __builtin_amdgcn_global_store_async_from_lds_b32(
        (fk_gi32p)(gout + i), (fk_li32p)(&lds[r * FK_XF + c]), 0, 0);
  }
#if __has_builtin(__builtin_amdgcn_s_wait_asynccnt)
  __builtin_amdgcn_s_wait_asynccnt(0);
#else
  asm volatile("s_wait_asynccnt 0" ::: "memory");
#endif
#else
  for (int i = tid; i < NOUT; i += FK_BLOCK) {
    const int r = i / FK_PF;
    const int c = i - r * FK_PF;
    gout[i] = lds[r * FK_XF + c];
  }
#endif
}

// Rows not covered by full tiles (none for B=262144; kept for robustness).
// pos goes straight to global; store->load same address same wave is ordered.
__global__ void fk_tail(const float* __restrict__ x, const float* __restrict__ off,
                        float* __restrict__ out, int start, int n) {
  int row = start + blockIdx.x * blockDim.x + threadIdx.x;
  if (row >= n) return;
  fk_row(x + (long long)row * FK_XF, off, out + (long long)row * FK_PF);
}

extern "C" void kernel_launch(void* const* d_in, const int* in_sizes, int n_in,
                              void* d_out, int out_size, void* d_ws, size_t ws_size,
                              hipStream_t stream) {
  const float* x = (const float*)d_in[0];
  const float* off = (const float*)d_in[1];
  float* out = (float*)d_out;
  const int n = in_sizes[0] / FK_XF;      // number of samples (262144)
  const int blocks = n / FK_BLOCK;        // full tiles
  if (blocks > 0)
    fk_kernel<<<blocks, FK_BLOCK, 0, stream>>>(x, off, out);
  const int start = blocks * FK_BLOCK;
  const int rem = n - start;
  if (rem > 0)
    fk_tail<<<(rem + 127) / 128, 128, 0, stream>>>(x, off, out, start, n);
}